// LearnerNet_60507499266582
// MI455X (gfx1250) — compile-verified
//
#include <hip/hip_runtime.h>

// Problem dims (fixed by reference): B=32, IN=784, HID=256, OUT=128, MH=32.
#define BATCH 32

typedef float v2f __attribute__((ext_vector_type(2)));
typedef float v8f __attribute__((ext_vector_type(8)));

// Single-instruction relu. fmaxf/fmed3 both get canonicalize-wrapped by the
// compiler (it can't prove WMMA outputs are canonical); inline asm is opaque
// to that pass and emits exactly one v_max_num_f32 with inline-constant 0.
__device__ __forceinline__ float relu1(float x) {
    float r;
    asm("v_max_num_f32 %0, 0, %1" : "=v"(r) : "v"(x));
    return r;
}

// ---------------------------------------------------------------------------
// Forward: vj = relu(vi @ W^T + bias). Also seeds the output accumulator
// (new_out = vj + corrections, corrections added later by meta_kernel).
// Tiny (<1% of FLOPs) -> simple per-(b,o) thread with float4 loads.
// ---------------------------------------------------------------------------
__global__ void fwd_kernel(const float* __restrict__ vi, const float* __restrict__ W,
                           const float* __restrict__ bias, float* __restrict__ vj,
                           float* __restrict__ acc, int I, int O) {
    int o = blockIdx.x * blockDim.x + threadIdx.x;
    int b = blockIdx.y;
    if (o >= O) return;
    const float4* xr = reinterpret_cast<const float4*>(vi + (size_t)b * I);
    const float4* wr = reinterpret_cast<const float4*>(W + (size_t)o * I);
    float s = bias[o];
    int n4 = I >> 2;
    for (int i = 0; i < n4; ++i) {
        float4 a = xr[i], w = wr[i];
        s = fmaf(a.x, w.x, s); s = fmaf(a.y, w.y, s);
        s = fmaf(a.z, w.z, s); s = fmaf(a.w, w.w, s);
    }
    float r = relu1(s);
    vj[b * O + o]  = r;
    acc[b * O + o] = r;
}

// ---------------------------------------------------------------------------
// Meta kernel: per (o, 16-wide i-tile), loop b=0..31. For 16 triples at a
// time, evaluate the 3->32->1 meta MLP with V_WMMA_F32_16X16X4_F32.
//
//   A [16x4]  : columns (vi[b,i], W[o,i], vj[b,o], 1.0)   <- bias folded in K
//   B [4x16]  : rows    (mW1[:,0], mW1[:,1], mW1[:,2], mb1) per 16-h chunk
//   D [16x16] : pre-activations for 16 triples x 16 hidden (x2 chunks = 32)
//
// f32 A layout (ISA 7.12.2): lanes 0-15 hold M=0..15 / K={0,1};
// lanes 16-31 hold M=0..15 / K={2,3}. A needs NO shuffles:
//   A.x = lane<16 ? vi : vj(broadcast);  A.y = lane<16 ? w : 1.0
//
// Second meta layer: p'[r] = relu(D0[r])*mW2a + relu(D1[r])*mW2b
//                           + (n==0 ? mb2 : 0)        (lane n = lane&15)
//   meta[t=r+8*hi] = sum_n p'[r]    -> weight update: accumulate p' over b
//                                      per-lane, reduce ONCE after the loop.
//   oc(b) = sum_t vi[t]*meta[t]     -> q = sum_r shfl(vi, r+8*hi)*p'[r],
//                                      5-step wave reduce, deposit in lane b;
//                                      ONE vector atomic after the loop.
//
// vj column preloaded per-lane (lane = batch), broadcast via v_readlane;
// vi tile load software-pipelined with an incrementally maintained uniform
// offset (no per-iteration 64-bit multiplies).
// ---------------------------------------------------------------------------
__global__ __launch_bounds__(128) void meta_kernel(
    const float* __restrict__ vi, const float* __restrict__ W,
    const float* __restrict__ vj, const float* __restrict__ mW1,
    const float* __restrict__ mb1, const float* __restrict__ mW2,
    const float* __restrict__ mb2, const int* __restrict__ batch_num,
    float* __restrict__ out_acc, float* __restrict__ W_new, int I, int O) {

    const int lane = threadIdx.x;            // 0..31 (wave32)
    const int o    = blockIdx.y * blockDim.y + threadIdx.y;
    const int i0   = blockIdx.x * 16;
    const int hi   = lane >> 4;              // lane half (selects K pair)
    const int ln   = lane & 15;
    const int b3   = (lane >> 3) & 1;

    const float scale = 1.0e-3f / (float)batch_num[0];   // RATE / batch_num

    // Loop-invariant meta weights in B-matrix layout. mW1 is [32,3] row-major.
    const int h0 = ln, h1 = ln + 16;
    v2f Bm0, Bm1;
    Bm0.x = hi ? mW1[3 * h0 + 2] : mW1[3 * h0 + 0];   // K2 : K0
    Bm0.y = hi ? mb1[h0]         : mW1[3 * h0 + 1];   // K3 : K1
    Bm1.x = hi ? mW1[3 * h1 + 2] : mW1[3 * h1 + 0];
    Bm1.y = hi ? mb1[h1]         : mW1[3 * h1 + 1];
    const float mw2a     = mW2[h0];          // second-layer weights, chunk 0
    const float mw2b     = mW2[h1];          // chunk 1
    const float mb2_lane = (ln == 0) ? mb2[0] : 0.0f;  // fold mb2 into p'(n=0)

    const float w_elem = W[(size_t)o * I + i0 + ln];  // lanes>=16 hold dup
    const float a1     = hi ? 1.0f : w_elem;          // A column K1/K3
    const float vjv    = vj[lane * O + o];   // lane = batch: whole vj column

    float wacc_p[8];                         // per-lane sum_b p'[r]
#pragma unroll
    for (int r = 0; r < 8; ++r) wacc_p[r] = 0.0f;
    float oacc = 0.0f;                       // lane b accumulates oc(b)

    const float* vip = vi + i0 + ln;         // per-lane base, row 0
    float  vi_nxt = vip[0];                  // pipelined tile load
    size_t voff   = 0;                       // uniform row offset (elements)
#pragma unroll 4
    for (int b = 0; b < BATCH; ++b) {
        const float vi_l = vi_nxt;
        voff += (b < BATCH - 1) ? (size_t)I : 0;   // scalar cselect+add
        vi_nxt = vip[voff];                        // prefetch next row
        // Uniform-index broadcast of vj[b,o] -> SGPR via v_readlane.
        const float vj_bo = __int_as_float(
            __builtin_amdgcn_readlane(__float_as_int(vjv), b));
        v2f A; A.x = hi ? vj_bo : vi_l; A.y = a1;

        v8f D0 = {}; v8f D1 = {};
        D0 = __builtin_amdgcn_wmma_f32_16x16x4_f32(false, A, false, Bm0,
                                                   (short)0, D0, false, false);
        D1 = __builtin_amdgcn_wmma_f32_16x16x4_f32(false, A, false, Bm1,
                                                   (short)0, D1, false, false);

        // p'[r]: this lane's (t=r+8*hi, n=ln) contribution to meta[t].
        // q: this lane's contribution to sum_t vi[t]*meta[t].
        float q = 0.0f;
#pragma unroll
        for (int r = 0; r < 8; ++r) {
            float p = fmaf(relu1(D0[r]), mw2a,
                           fmaf(relu1(D1[r]), mw2b, mb2_lane));
            wacc_p[r] += p;                          // deferred weight update
            float v_t = __shfl(vi_l, r + 8 * hi, 32);  // vi at triple t
            q = fmaf(v_t, p, q);
        }
        // Wave-wide sum of q == oc(b); deposit into lane b's accumulator.
        q += __shfl_xor(q, 1, 32);
        q += __shfl_xor(q, 2, 32);
        q += __shfl_xor(q, 4, 32);
        q += __shfl_xor(q, 8, 32);
        q += __shfl_xor(q, 16, 32);
        oacc += (lane == b) ? q : 0.0f;
    }

    // Output correction: one vector atomic, lane = batch index.
    atomicAdd(&out_acc[lane * O + o], oacc * scale);

    // Weight update: reduce wacc_p[r] over the 16 lanes of each half
    // (sum over n), giving sum_b meta[t=r+8*hi] replicated per half.
    float s[8];
#pragma unroll
    for (int r = 0; r < 8; ++r) {
        float p = wacc_p[r];
        p += __shfl_xor(p, 1, 32);
        p += __shfl_xor(p, 2, 32);
        p += __shfl_xor(p, 4, 32);
        p += __shfl_xor(p, 8, 32);
        s[r] = p;
    }
    // Land value for t = lane&15 in each lane: select s[lane&7] from own
    // half, swap with xor-24 partner where the half bits disagree.
    float t0 = (lane & 1) ? s[1] : s[0];
    float t1 = (lane & 1) ? s[3] : s[2];
    float t2 = (lane & 1) ? s[5] : s[4];
    float t3 = (lane & 1) ? s[7] : s[6];
    float u0 = (lane & 2) ? t1 : t0;
    float u1 = (lane & 2) ? t3 : t2;
    float cand = (lane & 4) ? u1 : u0;
    float part = __shfl_xor(cand, 24, 32);
    float wsum = (b3 != hi) ? part : cand;   // sum_b meta[t=lane&15]

    // W_new = W + mean_b(shift) = W + (scale/BATCH) * sum_b meta
    if (hi == 0) {
        W_new[(size_t)o * I + i0 + ln] = w_elem + wsum * (scale / (float)BATCH);
    }
}

// ---------------------------------------------------------------------------
// Host launch. Inputs: x, W1, b1, W2, b2, mW1, mb1, mW2, mb2, batch_num.
// Output: [out2 (32*128) | W1_new (256*784) | W2_new (128*256)] fp32.
// ---------------------------------------------------------------------------
extern "C" void kernel_launch(void* const* d_in, const int* in_sizes, int n_in,
                              void* d_out, int out_size, void* d_ws, size_t ws_size,
                              hipStream_t stream) {
    (void)in_sizes; (void)n_in; (void)out_size; (void)ws_size;

    const float* x   = (const float*)d_in[0];
    const float* W1  = (const float*)d_in[1];
    const float* b1  = (const float*)d_in[2];
    const float* W2  = (const float*)d_in[3];
    const float* b2  = (const float*)d_in[4];
    const float* mW1 = (const float*)d_in[5];
    const float* mb1 = (const float*)d_in[6];
    const float* mW2 = (const float*)d_in[7];
    const float* mb2 = (const float*)d_in[8];
    const int*   bn  = (const int*)d_in[9];

    const int B = 32, IN = 784, HID = 256, OUT = 128;

    float* out2   = (float*)d_out;                 // [B, OUT]
    float* W1_new = (float*)d_out + B * OUT;       // [HID, IN]
    float* W2_new = W1_new + HID * IN;             // [OUT, HID]

    float* vj1  = (float*)d_ws;                    // [B, HID] relu fwd L1
    float* acc1 = vj1 + B * HID;                   // [B, HID] out1 accumulator
    float* vj2  = acc1 + B * HID;                  // [B, OUT] relu fwd L2

    // Layer 1
    fwd_kernel<<<dim3((HID + 127) / 128, B), 128, 0, stream>>>(
        x, W1, b1, vj1, acc1, IN, HID);
    meta_kernel<<<dim3(IN / 16, HID / 4), dim3(32, 4), 0, stream>>>(
        x, W1, vj1, mW1, mb1, mW2, mb2, bn, acc1, W1_new, IN, HID);

    // Layer 2 (input = corrected out1; out2 region of d_out doubles as acc)
    fwd_kernel<<<dim3((OUT + 127) / 128, B), 128, 0, stream>>>(
        acc1, W2, b2, vj2, out2, HID, OUT);
    meta_kernel<<<dim3(HID / 16, OUT / 4), dim3(32, 4), 0, stream>>>(
        acc1, W2, vj2, mW1, mb1, mW2, mb2, bn, out2, W2_new, HID, OUT);
}